// MyGraphModel_38817914421629
// MI455X (gfx1250) — compile-verified
//
#include <hip/hip_runtime.h>

typedef __attribute__((ext_vector_type(2))) float v2f;
typedef __attribute__((ext_vector_type(8))) float v8f;

#define NN 100000
#define NE 1600000
#define IND 128
#define HIDD 128
#define OUTD 64

// ---------------- zero workspace ----------------
__global__ void sage_zero_f32(float* __restrict__ p, long long n) {
  long long i = (long long)blockIdx.x * blockDim.x + threadIdx.x;
  long long step = (long long)gridDim.x * blockDim.x;
  for (; i < n; i += step) p[i] = 0.0f;
}

// ---------------- degree count ----------------
__global__ void sage_degree(const int* __restrict__ dst, float* __restrict__ cnt, int nE) {
  int e = blockIdx.x * blockDim.x + threadIdx.x;
  if (e < nE) atomicAdd(&cnt[dst[e]], 1.0f);
}

// ---------------- edge scatter: sum[dst] += h[src]  (dim = 128) ----------------
__global__ void sage_scatter(const float* __restrict__ h,
                             const int* __restrict__ src,
                             const int* __restrict__ dst,
                             float* __restrict__ sum, int nE) {
  int gtid = blockIdx.x * blockDim.x + threadIdx.x;
  int edge = gtid >> 5;          // one wave32 per edge
  int lane = gtid & 31;          // 32 lanes x float4 = 128 floats
  if (edge >= nE) return;
  int s = src[edge];
  int d = dst[edge];
  float4 v = ((const float4*)(h + (long long)s * IND))[lane];
  float* sp = sum + (long long)d * IND + lane * 4;
  atomicAdd(sp + 0, v.x);
  atomicAdd(sp + 1, v.y);
  atomicAdd(sp + 2, v.z);
  atomicAdd(sp + 3, v.w);
}

// ---------------- fused SAGE layer: out = act( (sum/cnt)@Wl + b + x@Wr ) ----------------
// K = 128 (IND). One block = 16 node rows; one wave per 16-column tile of NOUT.
// blockDim.x = 32 * (NOUT/16).
template <int NOUT, bool RELU>
__global__ void sage_gemm(const float* __restrict__ sum, const float* __restrict__ cnt,
                          const float* __restrict__ x,
                          const float* __restrict__ Wl, const float* __restrict__ bias,
                          const float* __restrict__ Wr,
                          float* __restrict__ out, int nNodes) {
  __shared__ float As[16][IND];   // mean-aggregated neighbor features
  __shared__ float Xs[16][IND];   // root features
  const int tid = threadIdx.x;
  const int node0 = blockIdx.x * 16;

  // cooperative load of 16 rows x 128 floats (as float4), scaling aggr by 1/max(cnt,1)
  for (int idx = tid; idx < 16 * (IND / 4); idx += blockDim.x) {
    int row = idx >> 5;          // IND/4 == 32 float4 per row
    int k4  = idx & 31;
    int node = node0 + row;
    float4 a  = make_float4(0.f, 0.f, 0.f, 0.f);
    float4 xv = make_float4(0.f, 0.f, 0.f, 0.f);
    float inv = 1.0f;
    if (node < nNodes) {
      inv = 1.0f / fmaxf(cnt[node], 1.0f);
      a   = ((const float4*)(sum + (long long)node * IND))[k4];
      xv  = ((const float4*)(x   + (long long)node * IND))[k4];
    }
    float4 as = make_float4(a.x * inv, a.y * inv, a.z * inv, a.w * inv);
    ((float4*)&As[row][0])[k4] = as;
    ((float4*)&Xs[row][0])[k4] = xv;
  }
  __syncthreads();

  const int wave  = tid >> 5;
  const int lane  = tid & 31;
  const int n0    = wave * 16;          // this wave's 16-column tile
  const int mrow  = lane & 15;          // A fragment: row = lane % 16
  const int khalf = (lane >> 4) << 1;   // lanes 16-31 hold K+2,K+3
  const int ncol  = n0 + mrow;          // B/C/D fragment column

  v8f acc = {};

  // term 1: (sum/cnt) @ Wl
  for (int k0 = 0; k0 < IND; k0 += 4) {
    int kb = k0 + khalf;
    v2f a, b;
    a.x = As[mrow][kb];
    a.y = As[mrow][kb + 1];
    b.x = Wl[kb * NOUT + ncol];
    b.y = Wl[(kb + 1) * NOUT + ncol];
    acc = __builtin_amdgcn_wmma_f32_16x16x4_f32(false, a, false, b,
                                                (short)0, acc, false, false);
  }
  // term 2: x @ Wr
  for (int k0 = 0; k0 < IND; k0 += 4) {
    int kb = k0 + khalf;
    v2f a, b;
    a.x = Xs[mrow][kb];
    a.y = Xs[mrow][kb + 1];
    b.x = Wr[kb * NOUT + ncol];
    b.y = Wr[(kb + 1) * NOUT + ncol];
    acc = __builtin_amdgcn_wmma_f32_16x16x4_f32(false, a, false, b,
                                                (short)0, acc, false, false);
  }

  // epilogue: bias (+ReLU) and store. C/D layout: VGPR j -> row j (+8 for lanes>=16)
  float bv = bias[ncol];
  int rbase = (lane >> 4) << 3;
#pragma unroll
  for (int j = 0; j < 8; ++j) {
    int node = node0 + rbase + j;
    if (node < nNodes) {
      float v = acc[j] + bv;
      if (RELU) v = fmaxf(v, 0.0f);
      out[(long long)node * NOUT + ncol] = v;
    }
  }
}

extern "C" void kernel_launch(void* const* d_in, const int* in_sizes, int n_in,
                              void* d_out, int out_size, void* d_ws, size_t ws_size,
                              hipStream_t stream) {
  const float* x   = (const float*)d_in[0];
  const int*   ei  = (const int*)d_in[1];   // [2, NE] flattened
  const float* W1l = (const float*)d_in[2];
  const float* b1  = (const float*)d_in[3];
  const float* W1r = (const float*)d_in[4];
  const float* W2l = (const float*)d_in[5];
  const float* b2  = (const float*)d_in[6];
  const float* W2r = (const float*)d_in[7];
  float* out = (float*)d_out;

  const int* src = ei;
  const int* dst = ei + NE;

  // workspace layout: cnt [NN] | sum [NN*128] | h [NN*128]
  float* cnt = (float*)d_ws;
  float* sum = cnt + NN;
  float* h   = sum + (long long)NN * IND;

  const long long zero_n = (long long)NN * (1 + IND);  // cnt + sum (contiguous)

  // ---- layer 1 ----
  sage_zero_f32<<<4096, 256, 0, stream>>>(cnt, zero_n);
  sage_degree<<<(NE + 255) / 256, 256, 0, stream>>>(dst, cnt, NE);
  sage_scatter<<<(NE * 32 + 255) / 256, 256, 0, stream>>>(x, src, dst, sum, NE);
  sage_gemm<HIDD, true><<<NN / 16, 32 * (HIDD / 16), 0, stream>>>(
      sum, cnt, x, W1l, b1, W1r, h, NN);

  // ---- layer 2 (reuse sum buffer) ----
  sage_zero_f32<<<4096, 256, 0, stream>>>(sum, (long long)NN * IND);
  sage_scatter<<<(NE * 32 + 255) / 256, 256, 0, stream>>>(h, src, dst, sum, NE);
  sage_gemm<OUTD, false><<<NN / 16, 32 * (OUTD / 16), 0, stream>>>(
      sum, cnt, h, W2l, b2, W2r, out, NN);
}